// SAGE_23235773072076
// MI455X (gfx1250) — compile-verified
//
#include <hip/hip_runtime.h>
#include <hip/hip_bf16.h>

typedef __attribute__((ext_vector_type(2))) float v2f;
typedef __attribute__((ext_vector_type(8))) float v8f;

#define IN_FEATS 64
#define N_CLASSES 64

// ---------------------------------------------------------------------------
// Kernel 0: zero scratch (summed + deg). d_ws is poisoned, so re-zero each call.
// ---------------------------------------------------------------------------
__global__ void sage_zero_kernel(float* __restrict__ p, long long n) {
    long long i = (long long)blockIdx.x * blockDim.x + threadIdx.x;
    long long stride = (long long)gridDim.x * blockDim.x;
    for (; i < n; i += stride) p[i] = 0.0f;
}

// ---------------------------------------------------------------------------
// Kernel 1: edge aggregation. 16 lanes per edge, each lane owns 4 features.
// src/dst index loads are uniform across the 16 lanes (broadcast from cache),
// the x-row gather is a coalesced 256B read, scatter uses HW f32 atomics
// (x and summed are both L2-resident: 25.6 MB each vs 192 MB L2).
// ---------------------------------------------------------------------------
__global__ void sage_agg_kernel(const float* __restrict__ x,
                                const int* __restrict__ src,
                                const int* __restrict__ dst,
                                float* __restrict__ summed,
                                float* __restrict__ deg,
                                int n_edges) {
    int t = blockIdx.x * blockDim.x + threadIdx.x;
    int e = t >> 4;
    if (e >= n_edges) return;
    int c = (t & 15) << 2;  // feature chunk: 0,4,...,60
    int s = src[e];
    int d = dst[e];
    const float4 v = *(const float4*)(x + (long long)s * IN_FEATS + c);
    float* o = summed + (long long)d * IN_FEATS + c;
    unsafeAtomicAdd(o + 0, v.x);
    unsafeAtomicAdd(o + 1, v.y);
    unsafeAtomicAdd(o + 2, v.z);
    unsafeAtomicAdd(o + 3, v.w);
    if (c == 0) unsafeAtomicAdd(deg + d, 1.0f);
}

// ---------------------------------------------------------------------------
// Kernel 2: fused dual-GEMM + bias + relu via V_WMMA_F32_16X16X4_F32.
//   out[r, :] = relu( x[r,:] @ Wself + (summed[r,:]/max(deg[r],1)) @ Wneigh + b )
// Block = 256 threads = 8 waves. Wave w: colTile = w&3 (n0 = 16*colTile),
// rowGrp = w>>2; block covers 32 rows. One 16x16 f32 tile per wave.
// A-frag (16x4 f32): lanes 0-15 row M=lane K={0,1}; lanes 16-31 row M=lane-16 K={2,3}.
// B-frag (4x16 f32): lanes 0-15 col N=lane K={0,1}; lanes 16-31 col N=lane-16 K={2,3}.
// C/D (16x16 f32):   lanes 0-15 col N=lane rows M=vgpr 0..7; lanes 16-31 rows M=8..15.
// Two accumulators (self / neigh) keep two independent WMMA chains in flight.
// ---------------------------------------------------------------------------
__global__ void sage_gemm_kernel(const float* __restrict__ x,
                                 const float* __restrict__ Wself,
                                 const float* __restrict__ Wneigh,
                                 const float* __restrict__ bias,
                                 const float* __restrict__ summed,
                                 const float* __restrict__ deg,
                                 float* __restrict__ out,
                                 int num_rows) {
    const int lane    = threadIdx.x & 31;
    const int wave    = threadIdx.x >> 5;   // 0..7
    const int colTile = wave & 3;           // 0..3
    const int rowGrp  = wave >> 2;          // 0..1
    const int row0    = blockIdx.x * 32 + rowGrp * 16;
    const int n0      = colTile * 16;
    const int half    = lane >> 4;          // 0: K={0,1}/rows 0-7   1: K={2,3}/rows 8-15
    const int l       = lane & 15;

    if (row0 + 16 <= num_rows) {
        // ---- preload B fragments for both weight matrices (K = 16 chunks of 4)
        v2f bs[16], bn[16];
#pragma unroll
        for (int kc = 0; kc < 16; ++kc) {
            const int k = kc * 4 + half * 2;
            bs[kc].x = Wself[(k + 0) * N_CLASSES + n0 + l];
            bs[kc].y = Wself[(k + 1) * N_CLASSES + n0 + l];
            bn[kc].x = Wneigh[(k + 0) * N_CLASSES + n0 + l];
            bn[kc].y = Wneigh[(k + 1) * N_CLASSES + n0 + l];
        }

        const int r = row0 + l;               // this lane's A row
        // deg is integer-valued; HW v_rcp_f32 (~1 ulp) replaces the full IEEE
        // divide expansion (div_scale/div_fmas/div_fixup) the compiler emitted.
        const float inv = __builtin_amdgcn_rcpf(fmaxf(deg[r], 1.0f));
        const float* xr = x + (long long)r * IN_FEATS;
        const float* sr = summed + (long long)r * IN_FEATS;

        v8f acc_s = {0.f, 0.f, 0.f, 0.f, 0.f, 0.f, 0.f, 0.f};
        v8f acc_n = {0.f, 0.f, 0.f, 0.f, 0.f, 0.f, 0.f, 0.f};
#pragma unroll
        for (int kc = 0; kc < 16; ++kc) {
            const int k = kc * 4 + half * 2;
            v2f a;  a.x = xr[k];         a.y = xr[k + 1];
            v2f an; an.x = sr[k] * inv;  an.y = sr[k + 1] * inv;
            acc_s = __builtin_amdgcn_wmma_f32_16x16x4_f32(
                false, a,  false, bs[kc], (short)0, acc_s, false, false);
            acc_n = __builtin_amdgcn_wmma_f32_16x16x4_f32(
                false, an, false, bn[kc], (short)0, acc_n, false, false);
        }

        const float bv = bias[n0 + l];
#pragma unroll
        for (int j = 0; j < 8; ++j) {
            const int m = half * 8 + j;
            float v = acc_s[j] + acc_n[j] + bv;
            v = fmaxf(v, 0.0f);
            out[(long long)(row0 + m) * N_CLASSES + n0 + l] = v;
        }
    } else if (row0 < num_rows) {
        // scalar tail (never taken for N=100000, kept for robustness;
        // wave-uniform branch so the WMMA path always runs with EXEC all-1s)
        if (half == 0) {
            for (int r2 = row0; r2 < num_rows; ++r2) {
                const float inv2 = __builtin_amdgcn_rcpf(fmaxf(deg[r2], 1.0f));
                float acc = bias[n0 + l];
                for (int k = 0; k < IN_FEATS; ++k) {
                    acc += x[(long long)r2 * IN_FEATS + k] * Wself[k * N_CLASSES + n0 + l];
                    acc += summed[(long long)r2 * IN_FEATS + k] * inv2 *
                           Wneigh[k * N_CLASSES + n0 + l];
                }
                out[(long long)r2 * N_CLASSES + n0 + l] = fmaxf(acc, 0.0f);
            }
        }
    }
}

// ---------------------------------------------------------------------------
// Launch: inputs are (x, W_self, W_neigh, b, src, dst, num_dst).
// num_dst derived host-side as in_sizes[0]/64 (graph-capture safe).
// ---------------------------------------------------------------------------
extern "C" void kernel_launch(void* const* d_in, const int* in_sizes, int n_in,
                              void* d_out, int out_size, void* d_ws, size_t ws_size,
                              hipStream_t stream) {
    const float* x      = (const float*)d_in[0];
    const float* Wself  = (const float*)d_in[1];
    const float* Wneigh = (const float*)d_in[2];
    const float* bias   = (const float*)d_in[3];
    const int*   src    = (const int*)d_in[4];
    const int*   dst    = (const int*)d_in[5];

    const int num_dst = in_sizes[0] / IN_FEATS;   // 100000
    const int n_edges = in_sizes[4];              // 1000000

    float* summed = (float*)d_ws;                           // num_dst*64 floats
    float* deg    = summed + (long long)num_dst * IN_FEATS; // num_dst floats

    // 1) zero scratch
    const long long nz = (long long)num_dst * (IN_FEATS + 1);
    sage_zero_kernel<<<2048, 256, 0, stream>>>(summed, nz);

    // 2) edge aggregation: 16 lanes per edge
    const long long agg_threads = (long long)n_edges * 16;
    const int agg_blocks = (int)((agg_threads + 255) / 256);
    sage_agg_kernel<<<agg_blocks, 256, 0, stream>>>(x, src, dst, summed, deg, n_edges);

    // 3) fused dual-GEMM + bias + relu (32 rows per block)
    const int gemm_blocks = (num_dst + 31) / 32;
    sage_gemm_kernel<<<gemm_blocks, 256, 0, stream>>>(x, Wself, Wneigh, bias,
                                                      summed, deg, (float*)d_out,
                                                      num_dst);
}